// GCN_84052509983292
// MI455X (gfx1250) — compile-verified
//
#include <hip/hip_runtime.h>

// ---------------------------------------------------------------------------
// GCN forward for MI455X (gfx1250, wave32).
//  h = x @ W          -> V_WMMA_F32_16X16X4_F32 (f32-exact tensor path)
//  agg = scatter_add(h[src]*norm, dst) + h*deg_inv^2 + b
//  x = relu(batchnorm(agg))
//  out = meanpool_by_graph(x) @ Wc + bc
// ---------------------------------------------------------------------------

typedef __attribute__((ext_vector_type(2))) float v2f;
typedef __attribute__((ext_vector_type(8))) float v8f;

#define DIM 128
#define NGRAPH 512
#define NCLS 10

static __device__ __forceinline__ float frelu(float v) { return v > 0.0f ? v : 0.0f; }

// ---------------- generic fill ----------------
__global__ void k_fill(float* __restrict__ p, float v, int n) {
    int t = blockIdx.x * blockDim.x + threadIdx.x;
    if (t < n) p[t] = v;
}

// ---------------- degree: atomicAdd 1.0 at dst ----------------
__global__ void k_deg(const int* __restrict__ dst, float* __restrict__ deg, int E) {
    int e = blockIdx.x * blockDim.x + threadIdx.x;
    if (e < E) atomicAdd(&deg[dst[e]], 1.0f);
}

// ---------------- deg -> deg^-0.5 in place ----------------
__global__ void k_rsqrt(float* __restrict__ d, int n) {
    int t = blockIdx.x * blockDim.x + threadIdx.x;
    if (t < n) d[t] = rsqrtf(d[t]);
}

// ---------------- norm[e] = dinv[src]*dinv[dst] ----------------
__global__ void k_norm(const int* __restrict__ src, const int* __restrict__ dst,
                       const float* __restrict__ dinv, float* __restrict__ norm, int E) {
    int e = blockIdx.x * blockDim.x + threadIdx.x;
    if (e < E) norm[e] = dinv[src[e]] * dinv[dst[e]];
}

// ---------------- h = X @ W via WMMA f32 16x16x4 ----------------
// One wave -> 16 rows x 128 cols. A fragments preloaded (32 x v2f),
// 8 column tiles x 32 k-steps of V_WMMA_F32_16X16X4_F32.
__global__ __launch_bounds__(256) void k_gemm_wmma(
    const float* __restrict__ X, const float* __restrict__ W,
    float* __restrict__ H, int N)
{
    int gwave = (int)((blockIdx.x * blockDim.x + threadIdx.x) >> 5);
    int lane  = threadIdx.x & 31;
    int row0  = gwave * 16;
    if (row0 >= N) return;                    // wave-uniform: EXEC stays all-ones
    int half = lane >> 4;                     // 0: lanes 0-15, 1: lanes 16-31
    int lid  = lane & 15;

    // A 16x4 fragment layout: lane holds M=lid, K = 2*half + {0,1} (v2f)
    const float* xr = X + (size_t)(row0 + lid) * DIM + 2 * half;
    v2f a[32];
#pragma unroll
    for (int k = 0; k < 32; ++k) {
        a[k].x = xr[4 * k];
        a[k].y = xr[4 * k + 1];
    }

#pragma unroll 1
    for (int ct = 0; ct < 8; ++ct) {
        int col = ct * 16 + lid;
        v8f acc = {0.f, 0.f, 0.f, 0.f, 0.f, 0.f, 0.f, 0.f};
#pragma unroll
        for (int k = 0; k < 32; ++k) {
            // B 4x16 fragment: lane holds N=col, K = 2*half + {0,1}
            v2f bf;
            bf.x = W[(size_t)(4 * k + 2 * half)     * DIM + col];
            bf.y = W[(size_t)(4 * k + 2 * half + 1) * DIM + col];
            acc = __builtin_amdgcn_wmma_f32_16x16x4_f32(
                      false, a[k], false, bf, (short)0, acc, false, false);
        }
        // C/D 16x16: VGPR j -> row M = j + 8*half, N = col
        float* hr = H + (size_t)row0 * DIM + col;
#pragma unroll
        for (int j = 0; j < 8; ++j)
            hr[(size_t)(j + 8 * half) * DIM] = acc[j];
    }
}

// ---------------- agg = h * dinv^2 + bias (self loop + bias) ----------------
__global__ void k_selfinit(const float* __restrict__ H, const float* __restrict__ dinv,
                           const float* __restrict__ bias, float* __restrict__ agg, int N) {
    int t = blockIdx.x * blockDim.x + threadIdx.x;
    if (t >= N * DIM) return;
    int i = t >> 7, c = t & 127;
    float di = dinv[i];
    agg[t] = H[t] * di * di + bias[c];
}

// ---------------- edge scatter: agg[dst] += h[src]*norm ----------------
// one wave per edge; each lane moves one float4 (512B/edge, coalesced)
__global__ __launch_bounds__(256) void k_scatter(
    const float4* __restrict__ H4, const int* __restrict__ src,
    const int* __restrict__ dst, const float* __restrict__ norm,
    float* __restrict__ agg, int E)
{
    int e    = (int)((blockIdx.x * blockDim.x + threadIdx.x) >> 5);
    int lane = threadIdx.x & 31;
    if (e >= E) return;
    int   s = src[e];
    int   d = dst[e];
    float w = norm[e];
    float4 v = H4[(size_t)s * 32 + lane];
    float* out = agg + (size_t)d * DIM + lane * 4;
    atomicAdd(out + 0, v.x * w);
    atomicAdd(out + 1, v.y * w);
    atomicAdd(out + 2, v.z * w);
    atomicAdd(out + 3, v.w * w);
}

// ---------------- BN column statistics ----------------
__global__ __launch_bounds__(128) void k_bnstats(
    const float* __restrict__ X, float* __restrict__ sums,
    float* __restrict__ sumsq, int N)
{
    int c = threadIdx.x;                       // one column per thread (coalesced rows)
    float s = 0.f, s2 = 0.f;
    for (int r = blockIdx.x; r < N; r += gridDim.x) {
        float v = X[(size_t)r * DIM + c];
        s += v; s2 += v * v;
    }
    atomicAdd(&sums[c], s);
    atomicAdd(&sumsq[c], s2);
}

// ---------------- BN apply + ReLU ----------------
__global__ void k_bnapply(const float* __restrict__ agg, const float* __restrict__ sums,
                          const float* __restrict__ sumsq, const float* __restrict__ gamma,
                          const float* __restrict__ beta, float* __restrict__ Xo, int N) {
    int t = blockIdx.x * blockDim.x + threadIdx.x;
    if (t >= N * DIM) return;
    int c = t & 127;
    float invN = 1.0f / (float)N;
    float mu   = sums[c] * invN;
    float var  = sumsq[c] * invN - mu * mu;    // biased variance (matches reference)
    float sc   = rsqrtf(var + 1e-5f) * gamma[c];
    Xo[t] = frelu((agg[t] - mu) * sc + beta[c]);
}

// ---------------- graph mean-pool accumulation ----------------
__global__ void k_pool(const float* __restrict__ X, const int* __restrict__ batch,
                       float* __restrict__ pool, float* __restrict__ counts, int N) {
    int t = blockIdx.x * blockDim.x + threadIdx.x;
    if (t >= N * DIM) return;
    int i = t >> 7, c = t & 127;
    int g = batch[i];
    atomicAdd(&pool[(size_t)g * DIM + c], X[t]);
    if (c == 0) atomicAdd(&counts[g], 1.0f);
}

// ---------------- classifier: out = (pool/count) @ Wc + bc ----------------
__global__ void k_classify(const float* __restrict__ pool, const float* __restrict__ counts,
                           const float* __restrict__ Wc, const float* __restrict__ bc,
                           float* __restrict__ out) {
    int t = blockIdx.x * blockDim.x + threadIdx.x;
    if (t >= NGRAPH * NCLS) return;
    int g = t / NCLS, k = t % NCLS;
    float acc = 0.f;
    const float* pg = pool + (size_t)g * DIM;
#pragma unroll 4
    for (int c = 0; c < DIM; ++c) acc += pg[c] * Wc[c * NCLS + k];
    float cnt = counts[g];
    out[t] = acc / (cnt > 1.0f ? cnt : 1.0f) + bc[k];
}

// ---------------------------------------------------------------------------
static inline int cdiv(long long a, int b) { return (int)((a + b - 1) / b); }

extern "C" void kernel_launch(void* const* d_in, const int* in_sizes, int n_in,
                              void* d_out, int out_size, void* d_ws, size_t ws_size,
                              hipStream_t stream) {
    const float* x     = (const float*)d_in[0];
    const float* W     = (const float*)d_in[1];   // [4,128,128]
    const float* b     = (const float*)d_in[2];   // [4,128]
    const float* gamma = (const float*)d_in[3];   // [4,128]
    const float* beta  = (const float*)d_in[4];   // [4,128]
    const float* Wc    = (const float*)d_in[5];   // [128,10]
    const float* bc    = (const float*)d_in[6];   // [10]
    const int*   ei    = (const int*)d_in[7];     // [2,E]
    const int*   batch = (const int*)d_in[8];     // [N]
    float*       out   = (float*)d_out;

    const int N = in_sizes[0] / DIM;              // 50000
    const int E = in_sizes[7] / 2;                // 800000
    const int* src = ei;
    const int* dst = ei + E;

    // workspace carving (floats, 512B-aligned blocks)
    const size_t NF = (size_t)N * DIM;
    float* ws    = (float*)d_ws;
    float* h     = ws;                            // [N,128]
    float* agg   = h + NF;                        // [N,128]
    float* xa    = agg + NF;                      // [N,128]
    float* dinv  = xa + NF;                       // [N] (pad to 128-mult)
    size_t npad  = ((size_t)N + 127) & ~(size_t)127;
    float* norm  = dinv + npad;                   // [E]
    size_t epad  = ((size_t)E + 127) & ~(size_t)127;
    float* sums  = norm + epad;                   // [128]
    float* sumsq = sums + DIM;                    // [128]
    float* pool  = sumsq + DIM;                   // [512,128]
    float* cnts  = pool + (size_t)NGRAPH * DIM;   // [512]

    const int TB = 256;

    // --- degree / normalization coefficients ---
    k_fill<<<cdiv(N, TB), TB, 0, stream>>>(dinv, 1.0f, N);       // deg starts at 1 (self loop)
    k_deg<<<cdiv(E, TB), TB, 0, stream>>>(dst, dinv, E);
    k_rsqrt<<<cdiv(N, TB), TB, 0, stream>>>(dinv, N);
    k_norm<<<cdiv(E, TB), TB, 0, stream>>>(src, dst, dinv, norm, E);

    const int gemm_blocks    = cdiv((long long)(N / 16) * 32, TB);
    const int elem_blocks    = cdiv((long long)NF, TB);
    const int scatter_blocks = cdiv((long long)E * 32, TB);

    const float* xin = x;
    for (int l = 0; l < 4; ++l) {
        k_gemm_wmma<<<gemm_blocks, TB, 0, stream>>>(xin, W + (size_t)l * DIM * DIM, h, N);
        k_selfinit<<<elem_blocks, TB, 0, stream>>>(h, dinv, b + l * DIM, agg, N);
        k_scatter<<<scatter_blocks, TB, 0, stream>>>((const float4*)h, src, dst, norm, agg, E);
        k_fill<<<1, 256, 0, stream>>>(sums, 0.0f, 2 * DIM);      // sums + sumsq adjacent
        k_bnstats<<<512, 128, 0, stream>>>(agg, sums, sumsq, N);
        k_bnapply<<<elem_blocks, TB, 0, stream>>>(agg, sums, sumsq,
                                                  gamma + l * DIM, beta + l * DIM, xa, N);
        xin = xa;
    }

    // --- pool + classify ---
    k_fill<<<cdiv(NGRAPH * DIM + NGRAPH, TB), TB, 0, stream>>>(pool, 0.0f, NGRAPH * DIM + NGRAPH);
    k_pool<<<elem_blocks, TB, 0, stream>>>(xa, batch, pool, cnts, N);
    k_classify<<<cdiv(NGRAPH * NCLS, TB), TB, 0, stream>>>(pool, cnts, Wc, bc, out);
}